// AcrBertModel_79018808312044
// MI455X (gfx1250) — compile-verified
//
#include <hip/hip_runtime.h>
#include <math.h>

// CDNA5 (gfx1250) wave32 WMMA fragment types
typedef __attribute__((ext_vector_type(2))) float v2f;
typedef __attribute__((ext_vector_type(8))) float v8f;

#define BATCH 1024
#define STOK  512
#define HDIM  768
#define K2H   1536   // 2*H
#define D1    128
#define WAVES 4      // waves per block in the GEMM kernel
#define KT    64     // K-panel staged in LDS
#define SWP   144    // padded pair-row length (bank-shift +32 between pair rows)

// ---------------------------------------------------------------------------
// Kernel A: ragged span mean-pool + CLS concat.  One block per batch example.
// 192 threads * float4 = 768 floats per row, fully coalesced b128 loads.
// Only reads token 0 and tokens [start,end) (<=65 rows) per example.
// ---------------------------------------------------------------------------
__global__ __launch_bounds__(192) void pool_concat_kernel(
    const float* __restrict__ feat, const int* __restrict__ startI,
    const int* __restrict__ endI, float* __restrict__ X)
{
    const int b  = blockIdx.x;
    const int c4 = threadIdx.x;                 // float4 column, 192 per row
    const float4* base = reinterpret_cast<const float4*>(feat + (size_t)b * STOK * HDIM);
    const int s0 = startI[b];
    const int s1 = endI[b];

    float4 cls = base[c4];                      // token 0
    float4 acc = make_float4(0.f, 0.f, 0.f, 0.f);
    for (int s = s0; s < s1; ++s) {
        float4 v = base[(size_t)s * (HDIM / 4) + c4];
        if (s + 1 < s1)
            __builtin_prefetch(&base[(size_t)(s + 1) * (HDIM / 4) + c4], 0, 1);
        acc.x += v.x; acc.y += v.y; acc.z += v.z; acc.w += v.w;
    }
    const float inv = 1.0f / (float)(s1 - s0);  // span_len >= 1 guaranteed
    float4 crc = make_float4(acc.x * inv, acc.y * inv, acc.z * inv, acc.w * inv);

    float4* xrow = reinterpret_cast<float4*>(X + (size_t)b * K2H);
    xrow[c4] = cls;                             // [0,768): CLS
    xrow[(HDIM / 4) + c4] = crc;                // [768,1536): span mean
}

// ---------------------------------------------------------------------------
// Kernel B: h = relu(X*W1 + b1); out = sigmoid(h*W2 + b2)
// X:[1024,1536]  W1:[1536,128]  ->  V_WMMA_F32_16X16X4_F32, K-panels via LDS.
// W1 panel stored K-pair interleaved: sW[r][n] = {W1[k0+2r][n], W1[k0+2r+1][n]}
// so each B fragment is one aligned 8-byte ds load into an even VGPR pair.
// Each wave: one 16-row M-tile x all 8 N-tiles (8 x v8f accumulators).
// Epilogue folds the D1->1 projection with a __shfl_xor reduction.
// ---------------------------------------------------------------------------
__global__ __launch_bounds__(128) void mlp_head_kernel(
    const float* __restrict__ X, const float* __restrict__ W1,
    const float* __restrict__ b1, const float* __restrict__ W2,
    const float* __restrict__ b2, float* __restrict__ out)
{
    __shared__ __align__(16) float sW[KT / 2][SWP][2];  // 36864 B: K-pair interleaved W1 panel
    __shared__ __align__(16) float sX[WAVES][16][68];   // 17408 B: per-wave X panel, padded

    const int tid   = threadIdx.x;
    const int wave  = tid >> 5;
    const int lane  = tid & 31;
    const int half  = lane >> 4;     // selects K-pair (A/B), M-half (C/D)
    const int l     = lane & 15;
    const int mBase = (blockIdx.x * WAVES + wave) * 16;

    v8f acc[8];
#pragma unroll
    for (int nt = 0; nt < 8; ++nt) acc[nt] = (v8f){};

    for (int k0 = 0; k0 < K2H; k0 += KT) {
        // Stage W1[k0..k0+63][0..127] cooperatively, K-pair interleaved.
        // 32 pair-rows x 32 chunks (4 n each) = 1024 chunks, 8 per thread.
#pragma unroll
        for (int j = 0; j < 8; ++j) {
            int p  = tid + j * 128;          // chunk id
            int r  = p >> 5;                 // pair row 0..31
            int c  = (p & 31) * 4;           // n base (multiple of 4)
            float4 e = *reinterpret_cast<const float4*>(W1 + (size_t)(k0 + 2 * r + 0) * D1 + c);
            float4 o = *reinterpret_cast<const float4*>(W1 + (size_t)(k0 + 2 * r + 1) * D1 + c);
            *reinterpret_cast<float4*>(&sW[r][c + 0][0]) = make_float4(e.x, o.x, e.y, o.y);
            *reinterpret_cast<float4*>(&sW[r][c + 2][0]) = make_float4(e.z, o.z, e.w, o.w);
        }
        // Stage this wave's X rows [mBase..mBase+15][k0..k0+63]
#pragma unroll
        for (int j = 0; j < 8; ++j) {
            int p  = lane + j * 32;          // 256 float4 slots
            int r  = p >> 4;                 // 16 float4 per row
            int c  = (p & 15) * 4;
            float4 x = *reinterpret_cast<const float4*>(X + (size_t)(mBase + r) * K2H + k0 + c);
            *reinterpret_cast<float4*>(&sX[wave][r][c]) = x;
        }
        __syncthreads();

#pragma unroll 4
        for (int kk = 0; kk < KT; kk += 4) {
            // A 16x4 frag: lane half 0 -> K={kk,kk+1}, half 1 -> K={kk+2,kk+3}; M = l
            v2f a;
            a.x = sX[wave][l][kk + 2 * half + 0];
            a.y = sX[wave][l][kk + 2 * half + 1];
            const int pr = (kk >> 1) + half; // pair row: {K, K+1} interleaved
#pragma unroll
            for (int nt = 0; nt < 8; ++nt) {
                // B 4x16 frag as one aligned 8B LDS load (even VGPR pair)
                v2f bf = *reinterpret_cast<const v2f*>(&sW[pr][nt * 16 + l][0]);
                acc[nt] = __builtin_amdgcn_wmma_f32_16x16x4_f32(
                    false, a, false, bf, (short)0, acc[nt], false, false);
            }
        }
        __syncthreads();
    }

    // Epilogue: bias + ReLU, then fold W2 dot (D1 -> 1) in registers.
    // acc[nt][j] = h_pre[M = mBase + j + 8*half][N = nt*16 + l]
    float p[8];
#pragma unroll
    for (int j = 0; j < 8; ++j) p[j] = 0.0f;
#pragma unroll
    for (int nt = 0; nt < 8; ++nt) {
        const float bias = b1[nt * 16 + l];
        const float w2v  = W2[nt * 16 + l];
#pragma unroll
        for (int j = 0; j < 8; ++j) {
            float h = acc[nt][j] + bias;
            h = h > 0.0f ? h : 0.0f;
            p[j] += h * w2v;
        }
    }
    // Reduce across the 16 lanes of each half (xor masks 1,2,4,8 stay in-half)
#pragma unroll
    for (int j = 0; j < 8; ++j) {
        p[j] += __shfl_xor(p[j], 1, 32);
        p[j] += __shfl_xor(p[j], 2, 32);
        p[j] += __shfl_xor(p[j], 4, 32);
        p[j] += __shfl_xor(p[j], 8, 32);
    }
    if (l == 0) {
        const float bb = b2[0];
#pragma unroll
        for (int j = 0; j < 8; ++j) {
            float z = p[j] + bb;
            out[mBase + half * 8 + j] = 1.0f / (1.0f + __expf(-z));
        }
    }
}

// ---------------------------------------------------------------------------
extern "C" void kernel_launch(void* const* d_in, const int* in_sizes, int n_in,
                              void* d_out, int out_size, void* d_ws, size_t ws_size,
                              hipStream_t stream)
{
    (void)in_sizes; (void)n_in; (void)out_size; (void)ws_size;

    const float* feat = (const float*)d_in[0];  // [B,S,H]
    const int*   st   = (const int*)d_in[1];    // [B]
    const int*   en   = (const int*)d_in[2];    // [B]
    const float* W1   = (const float*)d_in[3];  // [1536,128]
    const float* b1   = (const float*)d_in[4];  // [128]
    const float* W2   = (const float*)d_in[5];  // [128,1]
    const float* b2   = (const float*)d_in[6];  // [1]
    float*       outp = (float*)d_out;          // [B]
    float*       X    = (float*)d_ws;           // [B,1536] concat scratch (6 MB)

    pool_concat_kernel<<<BATCH, 192, 0, stream>>>(feat, st, en, X);
    mlp_head_kernel<<<BATCH / (16 * WAVES), 128, 0, stream>>>(X, W1, b1, W2, b2, outp);
}